// LengthRegulator_10840497455833
// MI455X (gfx1250) — compile-verified
//
#include <hip/hip_runtime.h>

#define B_      16
#define L_      512
#define D_      512
#define MAXLEN_ 4096
#define ROW_BYTES_ (D_ * 4)      // 2048 bytes per row
#define WAVES_PER_BLOCK_ 8

typedef float v4f __attribute__((ext_vector_type(4)));

// ---------------------------------------------------------------------------
// Kernel 1: per-batch inclusive scan of durations + scatter of source indices.
// idx[b*MAXLEN + t] = l  such that cs[l-1] <= t < cs[l]   (t < total)
//                   = -1 for masked tail (t >= total)
// ---------------------------------------------------------------------------
__global__ __launch_bounds__(L_) void lr_build_idx(const int* __restrict__ dur,
                                                   int* __restrict__ idx) {
  __shared__ int cs[L_];
  const int b = blockIdx.x;
  const int l = threadIdx.x;

  cs[l] = dur[b * L_ + l];

  // Hillis-Steele inclusive scan over 512 entries (9 steps).
  for (int off = 1; off < L_; off <<= 1) {
    __syncthreads();
    int v = (l >= off) ? cs[l - off] : 0;
    __syncthreads();
    cs[l] += v;
  }
  __syncthreads();

  int end   = min(cs[l], MAXLEN_);
  int start = (l == 0) ? 0 : min(cs[l - 1], MAXLEN_);
  int* row = idx + (size_t)b * MAXLEN_;
  // durations are in [0,8) so this loop is at most 7 iterations
  for (int t = start; t < end; ++t) row[t] = l;

  // masked tail
  int total = min(cs[L_ - 1], MAXLEN_);
  for (int t = total + l; t < MAXLEN_; t += L_) row[t] = -1;
}

// ---------------------------------------------------------------------------
// Kernel 2: one 2KB output row per wave.
//   valid rows : global -> LDS -> global via gfx1250 async copy engine
//                (ASYNCcnt-tracked, non-temporal store to avoid polluting L2)
//   masked rows: non-temporal zero fill
// ---------------------------------------------------------------------------
__global__ __launch_bounds__(WAVES_PER_BLOCK_ * 32) void lr_expand(
    const float* __restrict__ x, const int* __restrict__ idx,
    float* __restrict__ out) {
  __shared__ unsigned char smem[WAVES_PER_BLOCK_ * ROW_BYTES_];

  const unsigned lane = threadIdx.x & 31u;
  const unsigned wave = threadIdx.x >> 5;
  const unsigned row  = blockIdx.x * WAVES_PER_BLOCK_ + wave;  // [0, B*MAXLEN)
  const unsigned b    = row >> 12;                             // row / 4096

  int src = idx[row];                       // same value across the wave
  src = __builtin_amdgcn_readfirstlane(src);

  float* outRow = out + (size_t)row * D_;

  if (src >= 0) {
    const float* xRow = x + ((size_t)b * L_ + (size_t)src) * D_;

    // low 32 bits of the generic pointer == addrspace(3) byte offset
    unsigned ldsBase = (unsigned)(unsigned long long)(uintptr_t)&smem[0];
    unsigned ldsOff  = ldsBase + wave * ROW_BYTES_ + lane * 16u;
    unsigned vOff    = lane * 16u;
    unsigned long long sa = (unsigned long long)(uintptr_t)xRow;
    unsigned long long da = (unsigned long long)(uintptr_t)outRow;

    // 4 x 512B async loads into LDS, wait, 4 x 512B async NT stores out.
    // The same immediate offset is applied to both the LDS VGPR address and
    // the global (SADDR + VADDR) address, so one pair of address VGPRs covers
    // the whole 2KB row.
    asm volatile(
        "global_load_async_to_lds_b128 %0, %1, %2 offset:0\n\t"
        "global_load_async_to_lds_b128 %0, %1, %2 offset:512\n\t"
        "global_load_async_to_lds_b128 %0, %1, %2 offset:1024\n\t"
        "global_load_async_to_lds_b128 %0, %1, %2 offset:1536\n\t"
        "s_wait_asynccnt 0\n\t"
        "global_store_async_from_lds_b128 %1, %0, %3 offset:0 th:TH_STORE_NT\n\t"
        "global_store_async_from_lds_b128 %1, %0, %3 offset:512 th:TH_STORE_NT\n\t"
        "global_store_async_from_lds_b128 %1, %0, %3 offset:1024 th:TH_STORE_NT\n\t"
        "global_store_async_from_lds_b128 %1, %0, %3 offset:1536 th:TH_STORE_NT\n\t"
        "s_wait_asynccnt 0"
        :
        : "v"(ldsOff), "v"(vOff), "s"(sa), "s"(da)
        : "memory");
  } else {
    // masked tail row: stream zeros, bypassing L2 residency
    v4f z = (v4f)(0.0f);
    v4f* o = (v4f*)outRow;
#pragma unroll
    for (int k = 0; k < 4; ++k) {
      __builtin_nontemporal_store(z, o + k * 32 + lane);
    }
  }
}

// ---------------------------------------------------------------------------
extern "C" void kernel_launch(void* const* d_in, const int* in_sizes, int n_in,
                              void* d_out, int out_size, void* d_ws,
                              size_t ws_size, hipStream_t stream) {
  const float* x   = (const float*)d_in[0];  // (B, L, D) f32
  const int*   dur = (const int*)d_in[1];    // (B, L) i32
  // d_in[2] = max_len scalar; MAXLEN_ is the compile-time constant 4096.
  (void)in_sizes; (void)n_in; (void)out_size; (void)ws_size;

  int*   idx = (int*)d_ws;                   // B*MAXLEN ints = 256 KB scratch
  float* out = (float*)d_out;                // (B, MAXLEN, D) f32

  lr_build_idx<<<B_, L_, 0, stream>>>(dur, idx);

  const int rows   = B_ * MAXLEN_;                       // 65536 rows
  const int blocks = rows / WAVES_PER_BLOCK_;            // 8192 blocks
  lr_expand<<<blocks, WAVES_PER_BLOCK_ * 32, 0, stream>>>(x, idx, out);
}